// AttentionHead_48284022342211
// MI455X (gfx1250) — compile-verified
//
#include <hip/hip_runtime.h>
#include <hip/hip_bf16.h>

#define NNODES 10000
#define DH     256
#define LEAKY  0.2f

typedef __attribute__((ext_vector_type(2))) float v2f;
typedef __attribute__((ext_vector_type(8))) float v8f;

// ---------------------------------------------------------------------------
// Kernel 1: feats = X @ W1 + b1 via V_WMMA_F32_16X16X4_F32.
// One wave32 per 16x16 output tile; 4 waves (128 threads) per block.
// Tiles: (10000/16)=625 M-tiles x 16 N-tiles = 10000 tiles = 2500 blocks.
// ---------------------------------------------------------------------------
__global__ __launch_bounds__(128)
void gat_gemm_wmma(const float* __restrict__ X, const float* __restrict__ W,
                   const float* __restrict__ bias, float* __restrict__ feats) {
    const int lane = threadIdx.x & 31;
    const int wave = threadIdx.x >> 5;
    const int tile = blockIdx.x * 4 + wave;      // 0..9999
    const int mt   = tile >> 4;                  // 0..624
    const int nt   = tile & 15;                  // 0..15
    const int row0 = mt * 16;
    const int col0 = nt * 16;
    const int half = lane >> 4;                  // 0: K+0/1, 1: K+2/3
    const int l    = lane & 15;

    const float* __restrict__ arow = X + (size_t)(row0 + l) * DH;  // A: row per lane
    const float* __restrict__ bcol = W + (col0 + l);               // B: column per lane

    v8f acc = {};
    #pragma unroll 8
    for (int k0 = 0; k0 < DH; k0 += 4) {
        const int kA = k0 + half * 2;
        v2f a, b;
        a.x = arow[kA];
        a.y = arow[kA + 1];
        b.x = bcol[(size_t)kA * DH];
        b.y = bcol[(size_t)(kA + 1) * DH];
        acc = __builtin_amdgcn_wmma_f32_16x16x4_f32(
            /*neg_a=*/false, a, /*neg_b=*/false, b,
            /*c_mod=*/(short)0, acc, /*reuse_a=*/false, /*reuse_b=*/false);
    }
    // C/D layout: VGPR r -> (M = r + 8*half, N = l)
    const float bv = bias[col0 + l];
    float* __restrict__ orow = feats + (size_t)(row0 + 8 * half) * DH + col0 + l;
    #pragma unroll
    for (int r = 0; r < 8; ++r)
        orow[(size_t)r * DH] = acc[r] + bv;
}

// ---------------------------------------------------------------------------
// Kernel 2: per-node attention dot products
//   s_src[i] = feats[i] . Wa[0:256],  s_dst[i] = feats[i] . Wa[256:512]
// One wave32 per node, 8 nodes per 256-thread block.
// ---------------------------------------------------------------------------
__global__ __launch_bounds__(256)
void gat_node_dots(const float* __restrict__ feats, const float* __restrict__ Wa,
                   float* __restrict__ s_src, float* __restrict__ s_dst) {
    const int lane = threadIdx.x & 31;
    const int node = blockIdx.x * 8 + (threadIdx.x >> 5);
    if (node >= NNODES) return;
    const float* __restrict__ f = feats + (size_t)node * DH;
    float a = 0.f, b = 0.f;
    #pragma unroll
    for (int t = lane; t < DH; t += 32) {
        const float v = f[t];
        a += v * Wa[t];
        b += v * Wa[DH + t];
    }
    #pragma unroll
    for (int off = 16; off > 0; off >>= 1) {
        a += __shfl_down(a, off, 32);
        b += __shfl_down(b, off, 32);
    }
    if (lane == 0) { s_src[node] = a; s_dst[node] = b; }
}

// ---------------------------------------------------------------------------
// Kernel 3: per-node edge softmax + weighted aggregation.
// src is sorted -> each node's edges are a contiguous range [lo, hi).
// One 256-thread block per node; thread t owns output feature t.
//   denom = sum exp(leaky(s_src[i] + s_dst[dst] + ba))
//   out[i,t] = (1/denom) * sum_e exp(..) * feats[dst_e, t]
// ---------------------------------------------------------------------------
__global__ __launch_bounds__(256)
void gat_aggregate(const float* __restrict__ feats, const long long* __restrict__ edges,
                   const float* __restrict__ s_src, const float* __restrict__ s_dst,
                   const float* __restrict__ ba, float* __restrict__ out, int nEdges) {
    const int i   = blockIdx.x;
    const int tid = threadIdx.x;

    __shared__ float exch[256];
    __shared__ int   dstch[256];
    __shared__ float red[256];
    __shared__ float s_inv;

    // Binary search: edge range with src == i (src = edges[2e], sorted).
    int lo, hiB;
    {
        int l = 0, r = nEdges;
        while (l < r) { int m = (l + r) >> 1; if (edges[2 * (size_t)m] < (long long)i) l = m + 1; else r = m; }
        lo = l;
        r = nEdges;
        while (l < r) { int m = (l + r) >> 1; if (edges[2 * (size_t)m] <= (long long)i) l = m + 1; else r = m; }
        hiB = l;
    }

    const float base = s_src[i] + ba[0];

    // Phase 1: softmax denominator (block reduction).
    float part = 0.f;
    for (int e = lo + tid; e < hiB; e += 256) {
        const int d = (int)edges[2 * (size_t)e + 1];
        float sc = base + s_dst[d];
        sc = sc > 0.f ? sc : LEAKY * sc;
        part += expf(sc);
    }
    red[tid] = part;
    __syncthreads();
    #pragma unroll
    for (int s = 128; s > 0; s >>= 1) {
        if (tid < s) red[tid] += red[tid + s];
        __syncthreads();
    }
    if (tid == 0) s_inv = (hiB > lo) ? 1.f / red[0] : 0.f;
    __syncthreads();
    const float inv = s_inv;

    // Phase 2: chunked weighted gather; feats rows hit in L2 (10 MB << 192 MB).
    float acc = 0.f;
    for (int c = lo; c < hiB; c += 256) {
        const int cnt = min(256, hiB - c);
        __syncthreads();
        if (tid < cnt) {
            const int e = c + tid;
            const int d = (int)edges[2 * (size_t)e + 1];
            float sc = base + s_dst[d];
            sc = sc > 0.f ? sc : LEAKY * sc;
            exch[tid]  = expf(sc);
            dstch[tid] = d;
        }
        __syncthreads();
        for (int j = 0; j < cnt; ++j)
            acc += exch[j] * feats[(size_t)dstch[j] * DH + tid];  // coalesced 1KB row reads
    }
    out[(size_t)i * DH + tid] = acc * inv;
}

// ---------------------------------------------------------------------------
// Launch. Inputs (setup_inputs order):
//   0: node_features f32 [10000,256]   1: edges int64 [320000,2]
//   2: W1 f32 [256,256]                3: b1 f32 [256]
//   4: Wa f32 [512,1]                  5: ba f32 [1]
// Output: f32 [10000,256]
// ---------------------------------------------------------------------------
extern "C" void kernel_launch(void* const* d_in, const int* in_sizes, int n_in,
                              void* d_out, int out_size, void* d_ws, size_t ws_size,
                              hipStream_t stream) {
    const float*     X     = (const float*)d_in[0];
    const long long* edges = (const long long*)d_in[1];   // reference: astype(jnp.int64)
    const float*     W1    = (const float*)d_in[2];
    const float*     b1    = (const float*)d_in[3];
    const float*     Wa    = (const float*)d_in[4];
    const float*     ba    = (const float*)d_in[5];
    float*           out   = (float*)d_out;

    const int nEdges = in_sizes[1] / 2;

    // Workspace partition
    float* feats = (float*)d_ws;                               // 10000*256 f32
    float* ssrc  = feats + (size_t)NNODES * DH;                // 10000 f32
    float* sdst  = ssrc + NNODES;                              // 10000 f32

    // 1) feats = X @ W1 + b1   (WMMA f32 16x16x4, one tile per wave)
    gat_gemm_wmma<<<(NNODES / 16) * 16 / 4, 128, 0, stream>>>(X, W1, b1, feats);

    // 2) per-node attention dots
    gat_node_dots<<<(NNODES + 7) / 8, 256, 0, stream>>>(feats, Wa, ssrc, sdst);

    // 3) per-node softmax + aggregation
    gat_aggregate<<<NNODES, 256, 0, stream>>>(feats, edges, ssrc, sdst, ba, out, nEdges);
}